// TemporalPrototypeManager_87986700026015
// MI455X (gfx1250) — compile-verified
//
#include <hip/hip_runtime.h>
#include <cstddef>

#define EPSN 1e-12f
#define MOM  0.9f
#define ONE_MINUS_MOM 0.1f

#define D 128
#define PAD 132            // 128 + 4 floats: kills LDS bank conflicts for strided WMMA reads
#define WPB 4              // waves per block
#define TPB (WPB * 32)

typedef float v2f __attribute__((ext_vector_type(2)));
typedef float v8f __attribute__((ext_vector_type(8)));

#if __has_builtin(__builtin_amdgcn_wmma_f32_16x16x4_f32)
#define HAVE_WMMA_F32X4 1
#else
#define HAVE_WMMA_F32X4 0
#endif

// ---------------------------------------------------------------------------
// Kernel 1: normalize rows + scatter-add into per-class segment sums.
// One wave processes a 16-row x 128-col tile:
//   - stage tile into LDS (coalesced float4 per lane per row)
//   - row sum-of-squares for all 16 rows via V_WMMA_F32_16X16X4_F32:
//       D = (A .* A) x ones(4x16), accumulated over K=128 in 32 steps.
//     Every column of D row m holds sumsq(row m).
//   - scatter normalized rows with f32 global atomics (L2-resident target)
// ---------------------------------------------------------------------------
__global__ __launch_bounds__(TPB) void norm_scatter_kernel(
    const float* __restrict__ feat, const int* __restrict__ labels,
    float* __restrict__ seg, float* __restrict__ cnt, int N)
{
    __shared__ __align__(16) float lds[WPB * 16 * PAD];

    const int lane = threadIdx.x & 31;
    const int wid  = threadIdx.x >> 5;
    float* wl = &lds[wid * 16 * PAD];

    const int gw    = blockIdx.x * WPB + wid;
    const int nw    = gridDim.x * WPB;
    const int tiles = (N + 15) >> 4;

    const int lrow = lane & 15;        // WMMA A row handled by this lane
    const int kof  = (lane >> 4) << 1; // K offset 0 (lanes 0-15) or 2 (lanes 16-31)

    for (int t = gw; t < tiles; t += nw) {
        const int row0 = t << 4;

        // ---- stage tile to LDS (uniform branch: row doesn't depend on lane)
        #pragma unroll
        for (int r = 0; r < 16; ++r) {
            const int row = row0 + r;
            float4 v = make_float4(0.f, 0.f, 0.f, 0.f);
            if (row < N)
                v = *(const float4*)(feat + (size_t)row * D + lane * 4);
            *(float4*)(wl + r * PAD + lane * 4) = v;
        }

        int lab = -1;
        if (lane < 16 && (row0 + lane) < N) lab = labels[row0 + lane];

#if HAVE_WMMA_F32X4
        // ---- 16 row-sumsqs on the matrix pipe
        v8f acc = {0.f, 0.f, 0.f, 0.f, 0.f, 0.f, 0.f, 0.f};
        v2f ones; ones[0] = 1.0f; ones[1] = 1.0f;
        const float* lp = wl + lrow * PAD + kof;
        #pragma unroll
        for (int kk = 0; kk < D; kk += 4) {
            const float x0 = lp[kk];
            const float x1 = lp[kk + 1];
            v2f a; a[0] = x0 * x0; a[1] = x1 * x1;
            acc = __builtin_amdgcn_wmma_f32_16x16x4_f32(
                false, a, false, ones, (short)0, acc, false, false);
        }
#endif

        // ---- scatter normalized rows (label uniform per wave per m)
        #pragma unroll
        for (int m = 0; m < 16; ++m) {
            const int labm = __shfl(lab, m, 32);
            if (labm < 0) continue;

#if HAVE_WMMA_F32X4
            const float ss = __shfl(acc[m & 7], (m < 8) ? 0 : 16, 32);
#else
            const float4 vv = *(const float4*)(wl + m * PAD + lane * 4);
            float ss = vv.x * vv.x + vv.y * vv.y + vv.z * vv.z + vv.w * vv.w;
            #pragma unroll
            for (int o = 16; o > 0; o >>= 1) ss += __shfl_xor(ss, o, 32);
#endif
            const float inv = 1.0f / fmaxf(sqrtf(ss), EPSN);

            const float4 v = *(const float4*)(wl + m * PAD + lane * 4);
            float* dst = seg + (size_t)labm * D + lane * 4;
            atomicAdd(dst + 0, v.x * inv);
            atomicAdd(dst + 1, v.y * inv);
            atomicAdd(dst + 2, v.z * inv);
            atomicAdd(dst + 3, v.w * inv);
            if (lane == 0) atomicAdd(cnt + labm, 1.0f);
        }
    }
}

// ---------------------------------------------------------------------------
// Kernel 2: per-class finalize. One wave (32 lanes x float4) per class.
// ---------------------------------------------------------------------------
__global__ __launch_bounds__(32) void finalize_kernel(
    const float* __restrict__ seg, const float* __restrict__ cnt,
    const float* __restrict__ proto, float* __restrict__ out, int C)
{
    const int c = blockIdx.x;
    if (c >= C) return;
    const int lane = threadIdx.x;

    const float4 s = *(const float4*)(seg   + (size_t)c * D + lane * 4);
    const float4 p = *(const float4*)(proto + (size_t)c * D + lane * 4);
    const float  n = cnt[c];
    const float den = fmaxf(n, 1.0f);

    float4 m = make_float4(s.x / den, s.y / den, s.z / den, s.w / den);

    // feat_mean norm
    float ssq = m.x * m.x + m.y * m.y + m.z * m.z + m.w * m.w;
    #pragma unroll
    for (int o = 16; o > 0; o >>= 1) ssq += __shfl_xor(ssq, o, 32);
    const float invm = 1.0f / fmaxf(sqrtf(ssq), EPSN);
    const float4 fm = make_float4(m.x * invm, m.y * invm, m.z * invm, m.w * invm);

    // prototypes elementwise sum (== 0 check, exact)
    float psum = p.x + p.y + p.z + p.w;
    #pragma unroll
    for (int o = 16; o > 0; o >>= 1) psum += __shfl_xor(psum, o, 32);

    // momentum EMA then norm
    float4 e = make_float4(MOM * p.x + ONE_MINUS_MOM * fm.x,
                           MOM * p.y + ONE_MINUS_MOM * fm.y,
                           MOM * p.z + ONE_MINUS_MOM * fm.z,
                           MOM * p.w + ONE_MINUS_MOM * fm.w);
    float esq = e.x * e.x + e.y * e.y + e.z * e.z + e.w * e.w;
    #pragma unroll
    for (int o = 16; o > 0; o >>= 1) esq += __shfl_xor(esq, o, 32);
    const float inve = 1.0f / fmaxf(sqrtf(esq), EPSN);
    const float4 en = make_float4(e.x * inve, e.y * inve, e.z * inve, e.w * inve);

    const bool isz = (psum == 0.0f);
    float4 np = isz ? fm : en;
    float4 res = (n > 0.0f) ? np : p;

    *(float4*)(out + (size_t)c * D + lane * 4) = res;
}

// ---------------------------------------------------------------------------
extern "C" void kernel_launch(void* const* d_in, const int* in_sizes, int n_in,
                              void* d_out, int out_size, void* d_ws, size_t ws_size,
                              hipStream_t stream)
{
    const float* feat   = (const float*)d_in[0];
    const int*   labels = (const int*)  d_in[1];
    const float* proto  = (const float*)d_in[2];
    float*       out    = (float*)d_out;

    const int N = in_sizes[0] / D;
    const int C = in_sizes[2] / D;

    float* seg = (float*)d_ws;          // [C, 128] segment sums
    float* cnt = seg + (size_t)C * D;   // [C] counts

    // zero accumulators every call (graph-capture-safe memset node)
    hipMemsetAsync(d_ws, 0, ((size_t)C * D + C) * sizeof(float), stream);

    const int tiles  = (N + 15) >> 4;
    int blocks = (tiles + WPB - 1) / WPB;
    if (blocks > 2048) blocks = 2048;
    if (blocks < 1) blocks = 1;

    norm_scatter_kernel<<<blocks, TPB, 0, stream>>>(feat, labels, seg, cnt, N);
    finalize_kernel<<<C, 32, 0, stream>>>(seg, cnt, proto, out, C);
}